// GradPooling_v2_63196148793925
// MI455X (gfx1250) — compile-verified
//
#include <hip/hip_runtime.h>
#include <cstdint>
#include <cstddef>

#ifndef __has_builtin
#define __has_builtin(x) 0
#endif

#if __has_builtin(__builtin_amdgcn_global_load_async_to_lds_b128)
#define GP_HAVE_ASYNC 1
#else
#define GP_HAVE_ASYNC 0
#endif

#if __has_builtin(__builtin_amdgcn_s_wait_asynccnt)
#define GP_HAVE_WAITASYNC 1
#else
#define GP_HAVE_WAITASYNC 0
#endif

namespace {
constexpr int kN  = 32;
constexpr int kH  = 56;
constexpr int kW  = 56;
constexpr int kC  = 256;
constexpr int kOH = 29;
constexpr int kOW = 29;
constexpr int kDR = 57;   // gradient-field extent with nonzero threshold weight
}

// 16-byte vector element type the async builtin expects, in the global (AS1)
// and LDS (AS3) address spaces.
typedef int gp_v4i __attribute__((ext_vector_type(4)));
typedef __attribute__((address_space(1))) gp_v4i gp_gv4i;
typedef __attribute__((address_space(3))) gp_v4i gp_lv4i;

// 16B global -> LDS copy; async data-mover path on CDNA5 (ASYNCcnt), plain
// load/store fallback if the builtin is absent on this toolchain.
__device__ __forceinline__ void gp_stage16(const float* gp, float* lp) {
#if GP_HAVE_ASYNC
  __builtin_amdgcn_global_load_async_to_lds_b128((gp_gv4i*)gp, (gp_lv4i*)lp, 0, 0);
#else
  const float4 v = *reinterpret_cast<const float4*>(gp);
  *reinterpret_cast<float4*>(lp) = v;
#endif
}

__device__ __forceinline__ void gp_wait_async() {
#if GP_HAVE_ASYNC
#if GP_HAVE_WAITASYNC
  __builtin_amdgcn_s_wait_asynccnt(0);
#else
  asm volatile("s_wait_asynccnt 0" ::: "memory");
#endif
#endif
}

// x~ with zero padding outside [0,56)^2, float4 over channels (NHWC, C fastest)
__device__ __forceinline__ float4 gp_load4(const float* __restrict__ x,
                                           int n, int r, int c, int ch4) {
  if ((unsigned)r < (unsigned)kH && (unsigned)c < (unsigned)kW) {
    const size_t idx = ((((size_t)n * kH + r) * kW + c) * kC) + (size_t)ch4 * 4;
    return *reinterpret_cast<const float4*>(x + idx);
  }
  return make_float4(0.f, 0.f, 0.f, 0.f);
}

// ---------------- Pass 1: weighted gradient sum partials -------------------
// block = (n, i) gradient row; 64 threads = float4 channel groups.
// d[i,j] = 2(|A-B| + |A-E|), A=x~(i,j), B=x~(i-2,j), E=x~(i,j-2);
// weight w(i)*w(j) with w(u)=2 (even u<=54), 1 (odd u, u==56).
__global__ void __launch_bounds__(64)
gp_diff_partial(const float* __restrict__ x, float* __restrict__ partial) {
  const int b = blockIdx.x;
  const int i = b % kDR;
  const int n = b / kDR;
  const int t = threadIdx.x;
  const float wi = (i == kDR - 1) ? 1.f : ((i & 1) ? 1.f : 2.f);
  float4 am2 = make_float4(0.f, 0.f, 0.f, 0.f);   // A at j-2
  float4 am1 = make_float4(0.f, 0.f, 0.f, 0.f);   // A at j-1
  float acc = 0.f;
  for (int j = 0; j < kDR; ++j) {
    const float4 a  = gp_load4(x, n, i,     j, t);
    const float4 bb = gp_load4(x, n, i - 2, j, t);
    const float4 e  = am2;
    const float wj = (j == kDR - 1) ? 1.f : ((j & 1) ? 1.f : 2.f);
    float s = fabsf(a.x - bb.x) + fabsf(a.x - e.x);
    s      += fabsf(a.y - bb.y) + fabsf(a.y - e.y);
    s      += fabsf(a.z - bb.z) + fabsf(a.z - e.z);
    s      += fabsf(a.w - bb.w) + fabsf(a.w - e.w);
    acc += wj * (2.f * s);
    am2 = am1;
    am1 = a;
  }
  acc *= wi;
  __shared__ float red[64];
  red[t] = acc;
  __syncthreads();
  for (int s = 32; s > 0; s >>= 1) {
    if (t < s) red[t] += red[t + s];
    __syncthreads();
  }
  if (t == 0) partial[b] = red[0];
}

// ---------------- Pass 2: deterministic final reduce -> threshold ----------
__global__ void __launch_bounds__(256)
gp_reduce_thresh(const float* __restrict__ partial, float* __restrict__ thresh,
                 int n_partial) {
  __shared__ float red[256];
  const int t = threadIdx.x;
  float acc = 0.f;
  for (int idx = t; idx < n_partial; idx += 256) acc += partial[idx];
  red[t] = acc;
  __syncthreads();
  for (int s = 128; s > 0; s >>= 1) {
    if (t < s) red[t] += red[t + s];
    __syncthreads();
  }
  if (t == 0) {
    const float denom = 9.f * kN * kC * kOH * kOW;  // mean over full dcol
    thresh[0] = red[0] / denom;
  }
}

// ---------------- Pass 3: fused gated pooling ------------------------------
// block = (n, oh, channel-group of 64). Stages x rows 2oh-3..2oh for 64
// channels into LDS via async global->LDS, then emits all 29 ow outputs.
__global__ void __launch_bounds__(256)
gp_pool(const float* __restrict__ x, const float* __restrict__ thresh_p,
        float* __restrict__ out) {
  constexpr int kRows = 4;
  constexpr int kChG  = 64;
  __shared__ float tile[kRows * kW * kChG];        // 57344 B

  const int b  = blockIdx.x;
  const int cg = b & 3;
  const int oh = (b >> 2) % kOH;
  const int n  = (b >> 2) / kOH;
  const int c0 = cg * kChG;
  const int rbase = 2 * oh - 3;
  const int t = threadIdx.x;

  // stage: 4 rows x 56 cols x 16 float4 chunks = 3584 chunks, 14 iters/thread.
  // Row-validity branch is warp-uniform (row changes every 896 chunks).
  for (int q = t; q < kRows * kW * (kChG / 4); q += 256) {
    const int cc4 = q & 15;
    const int w   = (q >> 4) % kW;
    const int r   = q / (16 * kW);
    const int gr  = rbase + r;
    float* lp = &tile[((r * kW + w) * kChG) + cc4 * 4];
    if ((unsigned)gr < (unsigned)kH) {
      const float* gp = x + ((((size_t)n * kH + gr) * kW + w) * kC) + c0 + cc4 * 4;
      gp_stage16(gp, lp);
    } else {
      lp[0] = 0.f; lp[1] = 0.f; lp[2] = 0.f; lp[3] = 0.f;   // padded rows
    }
  }
  gp_wait_async();
  __syncthreads();

  const float th = thresh_p[0];
  const int cc  = t & 63;
  const int ow0 = t >> 6;

  for (int ow = ow0; ow < kOW; ow += 4) {
    float sum = 0.f;
    float mx  = -3.4028235e38f;
    float ac  = 0.f;                       // tap (ky=1,kx=1) = x~(2oh-1,2ow-1)
#pragma unroll
    for (int ky = 0; ky < 3; ++ky) {
      const int ridx = ky + 1;             // rows 2oh-2 .. 2oh
#pragma unroll
      for (int kx = 0; kx < 3; ++kx) {
        const int g = 2 * ow - 2 + kx;
        float v = 0.f;
        if ((unsigned)g < (unsigned)kW) v = tile[(ridx * kW + g) * kChG + cc];
        sum += v;
        mx = fmaxf(mx, v);
        if (ky == 1 && kx == 1) ac = v;
      }
    }
    // center gradient: B = x~(2oh-3, 2ow-1), E = x~(2oh-1, 2ow-3)
    const int gB = 2 * ow - 1;
    const float Bv = ((unsigned)gB < (unsigned)kW) ? tile[(0 * kW + gB) * kChG + cc] : 0.f;
    const int gE = 2 * ow - 3;
    const float Ev = ((unsigned)gE < (unsigned)kW) ? tile[(2 * kW + gE) * kChG + cc] : 0.f;
    const float center = 2.f * (fabsf(ac - Bv) + fabsf(ac - Ev));
    const float res = (center > th) ? mx : (sum * (1.f / 9.f));
    out[((((size_t)n * kOH + oh) * kOW + ow) * kC) + c0 + cc] = res;
  }
}

extern "C" void kernel_launch(void* const* d_in, const int* in_sizes, int n_in,
                              void* d_out, int out_size, void* d_ws, size_t ws_size,
                              hipStream_t stream) {
  (void)in_sizes; (void)n_in; (void)out_size; (void)ws_size;
  const float* x = (const float*)d_in[0];
  float* out = (float*)d_out;
  float* partial = (float*)d_ws;       // kN*kDR = 1824 floats
  float* thresh  = partial + 2048;     // 1 float, past partials
  const int n_partial = kN * kDR;

  gp_diff_partial<<<n_partial, 64, 0, stream>>>(x, partial);
  gp_reduce_thresh<<<1, 256, 0, stream>>>(partial, thresh, n_partial);
  gp_pool<<<kN * kOH * 4, 256, 0, stream>>>(x, thresh, out);
}